// STARBodyModel_54331336294962
// MI455X (gfx1250) — compile-verified
//
#include <hip/hip_runtime.h>
#include <hip/hip_bf16.h>
#include <math.h>

// ---------------------------------------------------------------------------
// STAR/SMPL body model on MI455X (gfx1250, wave32).
// Both big GEMMs (posedirs blend-shape GEMM, LBS skinning GEMM) run on the
// fp32 WMMA pipe: v_wmma_f32_16x16x4_f32.
// ---------------------------------------------------------------------------

typedef __attribute__((ext_vector_type(2))) float v2f;
typedef __attribute__((ext_vector_type(8))) float v8f;

#define NB    1024          // batch
#define NJ    24            // joints
#define NV    6890          // vertices
#define NC    (NV * 3)      // 20670 flattened coords
#define NPAD  20672         // NC padded to multiple of 64 for GEMM tiles
#define KDIM  217           // 10 (beta) + 207 (pose feature)
#define KPAD  224           // padded to multiple of 4

// workspace layout (in floats)
#define AF_OFF   0
#define AF_SZ    ((size_t)NB * KPAD)                 //   229,376
#define BTP_OFF  (AF_OFF + AF_SZ)
#define BTP_SZ   ((size_t)KPAD * NPAD)               // 4,630,528
#define VP_OFF   (BTP_OFF + BTP_SZ)
#define VP_SZ    ((size_t)NB * NC)                   // 21,166,080
#define R_OFF    (VP_OFF + VP_SZ)
#define R_SZ     ((size_t)NB * NJ * 9)
#define GCP_OFF  (R_OFF + R_SZ)
#define GCP_SZ   ((size_t)NB * NJ * 16)              // pair-interleaved 24x16
// total ~26.64M floats = ~107 MB of d_ws

// ---------------------------------------------------------------------------
// Kernel 1: Rodrigues + assemble A = [beta | (R[1:]-I) | zero-pad]  (NB x KPAD)
// ---------------------------------------------------------------------------
__global__ void rodrigues_kernel(const float* __restrict__ theta,
                                 const float* __restrict__ beta,
                                 float* __restrict__ Rall,
                                 float* __restrict__ Afull)
{
    int id = blockIdx.x * blockDim.x + threadIdx.x;
    if (id >= NB * NJ) return;
    int b = id / NJ, j = id % NJ;

    float rx = theta[b * 72 + 3 * j + 0];
    float ry = theta[b * 72 + 3 * j + 1];
    float rz = theta[b * 72 + 3 * j + 2];
    float a  = sqrtf(rx * rx + ry * ry + rz * rz);
    a = fmaxf(a, 1e-8f);
    float inv = 1.0f / a;
    float x = rx * inv, y = ry * inv, z = rz * inv;
    float c = cosf(a), s = sinf(a), t = 1.0f - c;

    float R[9];
    R[0] = t * x * x + c;     R[1] = t * x * y - s * z; R[2] = t * x * z + s * y;
    R[3] = t * x * y + s * z; R[4] = t * y * y + c;     R[5] = t * y * z - s * x;
    R[6] = t * x * z - s * y; R[7] = t * y * z + s * x; R[8] = t * z * z + c;

    float* ro = Rall + (size_t)id * 9;
#pragma unroll
    for (int e = 0; e < 9; ++e) ro[e] = R[e];

    if (j > 0) {
        float* af = Afull + (size_t)b * KPAD + 10 + (j - 1) * 9;
#pragma unroll
        for (int e = 0; e < 9; ++e)
            af[e] = R[e] - ((e == 0 || e == 4 || e == 8) ? 1.0f : 0.0f);
    } else {
        float* af = Afull + (size_t)b * KPAD;
#pragma unroll
        for (int q = 0; q < 10; ++q) af[q] = beta[b * 10 + q];
        for (int q = KDIM; q < KPAD; ++q) af[q] = 0.0f;
    }
}

// ---------------------------------------------------------------------------
// Kernel 2: build pair-interleaved transposed B matrix Btp[(k>>1)][n][k&1]
//   B[k][n] = k<10 ? shapedirs[n][k] : (k<217 ? posedirs[n][k-10] : 0)
// so a WMMA B fragment (rows K,K+1 at column n) is one b64 load.
// ---------------------------------------------------------------------------
__global__ void build_btp_kernel(const float* __restrict__ sd,
                                 const float* __restrict__ pd,
                                 float* __restrict__ Btp)
{
    size_t id = (size_t)blockIdx.x * blockDim.x + threadIdx.x;
    if (id >= (size_t)KPAD * NPAD) return;
    int n = (int)(id % NPAD);
    int k = (int)(id / NPAD);
    float v = 0.0f;
    if (n < NC) {
        if (k < 10)        v = sd[(size_t)n * 10 + k];
        else if (k < KDIM) v = pd[(size_t)n * 207 + (k - 10)];
    }
    Btp[(((size_t)(k >> 1)) * NPAD + n) * 2 + (k & 1)] = v;
}

// ---------------------------------------------------------------------------
// Kernel 3: v_posed = A(NBxKPAD) @ B(KPADxNPAD) + v_template broadcast.
// One wave -> 16(batch) x 64(coord) tile: 4 fp32-WMMA accumulators, 56 K-chunks.
// ---------------------------------------------------------------------------
__global__ void __launch_bounds__(128)
vposed_gemm_kernel(const float* __restrict__ A,
                   const float* __restrict__ Btp,
                   const float* __restrict__ vt,
                   float* __restrict__ vp)
{
    const int lane = threadIdx.x & 31;
    const int wave = threadIdx.x >> 5;
    const int m    = lane & 15;     // A row within tile / B column within tile
    const int g    = lane >> 4;     // lane group (selects K pair)
    const int m0   = blockIdx.y * 16;
    const int slot = blockIdx.x * 4 + wave;      // 64-wide N slot
    if (slot >= NPAD / 64) return;               // whole-wave uniform (EXEC all-1)
    const int n0 = slot * 64;

    v8f c0 = {}, c1 = {}, c2 = {}, c3 = {};
    const float* arow = A + (size_t)(m0 + m) * KPAD;

#pragma unroll 4
    for (int kc = 0; kc < KPAD / 4; ++kc) {
        const int K = kc * 4 + 2 * g;                        // even
        v2f a = *(const v2f*)(arow + K);
        const float* bb = Btp + ((size_t)(kc * 2 + g) * NPAD + n0 + m) * 2;
        v2f b0 = *(const v2f*)(bb);
        v2f b1 = *(const v2f*)(bb + 32);
        v2f b2 = *(const v2f*)(bb + 64);
        v2f b3 = *(const v2f*)(bb + 96);
        if (kc + 4 < KPAD / 4)                               // global_prefetch_b8
            __builtin_prefetch(bb + (size_t)16 * NPAD, 0, 0);
        c0 = __builtin_amdgcn_wmma_f32_16x16x4_f32(false, a, false, b0, (short)0, c0, false, false);
        c1 = __builtin_amdgcn_wmma_f32_16x16x4_f32(false, a, false, b1, (short)0, c1, false, false);
        c2 = __builtin_amdgcn_wmma_f32_16x16x4_f32(false, a, false, b2, (short)0, c2, false, false);
        c3 = __builtin_amdgcn_wmma_f32_16x16x4_f32(false, a, false, b3, (short)0, c3, false, false);
    }

    // epilogue: add v_template broadcast and store (rows M = r + 8*g)
    v8f acc[4] = {c0, c1, c2, c3};
#pragma unroll
    for (int t = 0; t < 4; ++t) {
        int n = n0 + t * 16 + m;
        if (n < NC) {
            float add = vt[n];
#pragma unroll
            for (int r = 0; r < 8; ++r)
                vp[(size_t)(m0 + r + 8 * g) * NC + n] = acc[t][r] + add;
        }
    }
}

// ---------------------------------------------------------------------------
// Kernel 4: J[b][j][k] = sum_v Jreg[j][v] * v_posed[b][v][k]   (tree reduction)
// ---------------------------------------------------------------------------
__global__ void __launch_bounds__(256)
jreg_kernel(const float* __restrict__ Jreg,
            const float* __restrict__ vp,
            float* __restrict__ Jout)
{
    int bj = blockIdx.x;
    int b = bj / NJ, j = bj % NJ;
    const float* w    = Jreg + (size_t)j * NV;
    const float* vrow = vp + (size_t)b * NC;

    float a0 = 0.f, a1 = 0.f, a2 = 0.f;
    for (int v = threadIdx.x; v < NV; v += 256) {
        float wv = w[v];
        a0 += wv * vrow[3 * v + 0];
        a1 += wv * vrow[3 * v + 1];
        a2 += wv * vrow[3 * v + 2];
    }
    __shared__ float s0[256], s1[256], s2[256];
    int tid = threadIdx.x;
    s0[tid] = a0; s1[tid] = a1; s2[tid] = a2;
    __syncthreads();
    for (int off = 128; off > 0; off >>= 1) {
        if (tid < off) {
            s0[tid] += s0[tid + off];
            s1[tid] += s1[tid + off];
            s2[tid] += s2[tid + off];
        }
        __syncthreads();
    }
    if (tid == 0) {
        Jout[(size_t)bj * 3 + 0] = s0[0];
        Jout[(size_t)bj * 3 + 1] = s1[0];
        Jout[(size_t)bj * 3 + 2] = s2[0];
    }
}

// ---------------------------------------------------------------------------
// Kernel 5: kinematic chain -> G_corr, stored pair-interleaved over joints
//   Gcp[b][ (j>>1)*16 + c ][ j&1 ], c = k*4+l component of [R | t-R*J], c>=12 -> 0
// ---------------------------------------------------------------------------
__global__ void __launch_bounds__(32)
chain_kernel(const float* __restrict__ Rall,
             const float* __restrict__ Jv,
             float* __restrict__ Gcp)
{
    __shared__ float Gsh[32][NJ * 12];     // 36 KB of the 320 KB WGP LDS
    const int par[NJ] = {-1,0,0,0,1,2,3,4,5,6,7,8,9,9,9,12,13,14,16,17,18,19,20,21};
    int tid = threadIdx.x;
    int b = blockIdx.x * 32 + tid;
    if (b >= NB) return;
    float* outb = Gcp + (size_t)b * (NJ * 16);

    for (int j = 0; j < NJ; ++j) {
        float R[9], Jj[3], tr[3], Gj[12];
        const float* rp = Rall + ((size_t)b * NJ + j) * 9;
#pragma unroll
        for (int e = 0; e < 9; ++e) R[e] = rp[e];
#pragma unroll
        for (int k = 0; k < 3; ++k) Jj[k] = Jv[((size_t)b * NJ + j) * 3 + k];

        if (j == 0) {
#pragma unroll
            for (int k = 0; k < 3; ++k) tr[k] = Jj[k];
#pragma unroll
            for (int r = 0; r < 3; ++r) {
                Gj[r * 4 + 0] = R[r * 3 + 0];
                Gj[r * 4 + 1] = R[r * 3 + 1];
                Gj[r * 4 + 2] = R[r * 3 + 2];
                Gj[r * 4 + 3] = tr[r];
            }
        } else {
            float Jp[3];
#pragma unroll
            for (int k = 0; k < 3; ++k) Jp[k] = Jv[((size_t)b * NJ + par[j]) * 3 + k];
#pragma unroll
            for (int k = 0; k < 3; ++k)
                tr[k] = Jj[k] - (R[k*3+0]*Jp[0] + R[k*3+1]*Jp[1] + R[k*3+2]*Jp[2]);
            const float* Gp = Gsh[tid] + par[j] * 12;
#pragma unroll
            for (int r = 0; r < 3; ++r) {
#pragma unroll
                for (int c2 = 0; c2 < 3; ++c2)
                    Gj[r*4+c2] = Gp[r*4+0]*R[0*3+c2] + Gp[r*4+1]*R[1*3+c2] + Gp[r*4+2]*R[2*3+c2];
                Gj[r*4+3] = Gp[r*4+0]*tr[0] + Gp[r*4+1]*tr[1] + Gp[r*4+2]*tr[2] + Gp[r*4+3];
            }
        }
        float* gout = Gsh[tid] + j * 12;
#pragma unroll
        for (int e = 0; e < 12; ++e) gout[e] = Gj[e];

        float gt[3];
#pragma unroll
        for (int r = 0; r < 3; ++r)
            gt[r] = Gj[r*4+3] - (Gj[r*4+0]*Jj[0] + Gj[r*4+1]*Jj[1] + Gj[r*4+2]*Jj[2]);

#pragma unroll
        for (int c = 0; c < 16; ++c) {
            float v = 0.0f;
            if (c < 12) {
                int k = c >> 2, l = c & 3;
                v = (l < 3) ? Gj[k * 4 + l] : gt[k];
            }
            outb[((size_t)(j >> 1) * 16 + c) * 2 + (j & 1)] = v;
        }
    }
}

// ---------------------------------------------------------------------------
// Kernel 6: skinning. One wave -> 16-vertex tile.
//   T_tile(16x16, 12 used) = weights(16x24) @ Gcp[b](24x16)  via 6 fp32-WMMA,
//   stage tile in LDS, lanes 0..15 apply 3x4 transform to homogeneous v_posed.
// ---------------------------------------------------------------------------
__global__ void __launch_bounds__(128)
skin_kernel(const float* __restrict__ W,
            const float* __restrict__ Gcp,
            const float* __restrict__ vp,
            float* __restrict__ vout)
{
    __shared__ float lds[4][256];
    const int lane = threadIdx.x & 31;
    const int wave = threadIdx.x >> 5;
    const int m = lane & 15, g = lane >> 4;
    const int b = blockIdx.y;
    const int tile = blockIdx.x * 4 + wave;
    const bool active = (tile * 16) < NV;        // wave-uniform

    if (active) {
        const int v0 = tile * 16;
        const float* gb = Gcp + (size_t)b * (NJ * 16);
        v8f acc = {};
#pragma unroll
        for (int kc = 0; kc < 6; ++kc) {         // K = 24 joints = 6 chunks of 4
            const int K = kc * 4 + 2 * g;
            const int vrow = v0 + m;
            v2f a;
            if (vrow < NV) a = *(const v2f*)(W + (size_t)vrow * NJ + K);
            else           a = (v2f){0.0f, 0.0f};
            v2f bf = *(const v2f*)(gb + ((size_t)(kc * 2 + g) * 16 + m) * 2);
            acc = __builtin_amdgcn_wmma_f32_16x16x4_f32(false, a, false, bf, (short)0, acc, false, false);
        }
#pragma unroll
        for (int r = 0; r < 8; ++r)
            lds[wave][(r + 8 * g) * 16 + m] = acc[r];
    }
    __syncthreads();

    if (active && lane < 16) {
        const int vtx = tile * 16 + lane;
        if (vtx < NV) {
            const float* Ts = &lds[wave][lane * 16];
            const float* p  = vp + (size_t)b * NC + 3 * vtx;
            float x = p[0], y = p[1], z = p[2];
            float* o = vout + (size_t)b * NC + 3 * vtx;
#pragma unroll
            for (int k = 0; k < 3; ++k)
                o[k] = Ts[4*k+0]*x + Ts[4*k+1]*y + Ts[4*k+2]*z + Ts[4*k+3];
        }
    }
}

// ---------------------------------------------------------------------------
extern "C" void kernel_launch(void* const* d_in, const int* in_sizes, int n_in,
                              void* d_out, int out_size, void* d_ws, size_t ws_size,
                              hipStream_t stream)
{
    const float* theta      = (const float*)d_in[0];
    const float* beta       = (const float*)d_in[1];
    const float* shapedirs  = (const float*)d_in[2];
    const float* posedirs   = (const float*)d_in[3];
    const float* Jreg       = (const float*)d_in[4];
    const float* v_template = (const float*)d_in[5];
    const float* weights    = (const float*)d_in[6];

    float* vout = (float*)d_out;                       // (B, V, 3)
    float* Jout = (float*)d_out + (size_t)NB * NC;     // (B, 24, 3)

    float* ws     = (float*)d_ws;                      // needs ~107 MB
    float* Afull  = ws + AF_OFF;
    float* Btp    = ws + BTP_OFF;
    float* vposed = ws + VP_OFF;
    float* Rall   = ws + R_OFF;
    float* Gcp    = ws + GCP_OFF;

    // 1) rotations + pose-feature / beta assembly
    rodrigues_kernel<<<(NB * NJ + 255) / 256, 256, 0, stream>>>(theta, beta, Rall, Afull);

    // 2) pair-interleaved transposed blend-shape matrix
    build_btp_kernel<<<(unsigned)((KPAD * (size_t)NPAD) / 256), 256, 0, stream>>>(
        shapedirs, posedirs, Btp);

    // 3) main WMMA GEMM -> v_posed     (323 wave-slots of 64 cols, 64 batch tiles)
    vposed_gemm_kernel<<<dim3(81, 64), 128, 0, stream>>>(Afull, Btp, v_template, vposed);

    // 4) joint regressor reduction -> J (second output)
    jreg_kernel<<<NB * NJ, 256, 0, stream>>>(Jreg, vposed, Jout);

    // 5) kinematic chain -> pair-interleaved G_corr
    chain_kernel<<<32, 32, 0, stream>>>(Rall, Jout, Gcp);

    // 6) WMMA skinning -> v_out        (431 vertex tiles, 4 waves/block)
    skin_kernel<<<dim3(108, NB), 128, 0, stream>>>(weights, Gcp, vposed, vout);
}